// CRTNModel_85718957293865
// MI455X (gfx1250) — compile-verified
//
#include <hip/hip_runtime.h>

// CRTN 'middle_l' query construction (farnear path) — pure gather/broadcast.
//
//   out[i, j, b, h] = query_base[i + j + 1][b][h]
//   query_base      = concat(neighbor_mem[12], wise_inputs)  along axis 0
//
// Shapes (fp32):
//   neighbor_mem : (13, 128, 16, 1024)
//   wise_inputs  : (128, 16, 1024)
//   out          : (128, 128, 16, 1024)   = 1 GiB
//
// Store-bandwidth-bound: ~47 us floor at 23.3 TB/s. The 16 MiB gather base
// stays resident in the 192 MB L2 (RT loads); the 1 GiB output stream uses
// non-temporal stores (th:TH_STORE_NT) so it doesn't evict the base.
//
// This revision merges the two previous wins:
//  - base select done as integer math  -> single s_cselect_b64 (no branches,
//    round-2-style minimal SALU preamble)
//  - final addresses cast to address_space(1) pointers -> global_load_b128 /
//    global_store_b128 codegen (LOADcnt/STOREcnt only; no flat/DScnt coupling)

typedef float v4f __attribute__((ext_vector_type(4)));
typedef v4f __attribute__((address_space(1))) gv4f;   // global-memory v4f

#define CRTN_S        128
#define CRTN_B        16
#define CRTN_H        1024
#define CRTN_NEI      128
#define CRTN_LAYERS   13
#define ROW_FLOATS    (CRTN_B * CRTN_H)            // 16384 floats = 64 KB per row
#define ROW_VEC4      (ROW_FLOATS / 4)             // 4096 float4 per row
#define ROW_BYTES     ((uintptr_t)ROW_FLOATS * 4)  // 65536 bytes per row
#define BLOCK_THREADS 256                          // 8 wave32 per block
#define VEC_PER_THREAD (ROW_VEC4 / BLOCK_THREADS)  // 16

__global__ __launch_bounds__(BLOCK_THREADS)
void crtn_query_gather(const v4f* __restrict__ prev_,   // neighbor_mem[12], (128, 16384)
                       const v4f* __restrict__ wise_,   // wise_inputs,      (128, 16384)
                       v4f* __restrict__ out_)          // (16384, 16384)
{
    const unsigned row = blockIdx.x;          // row = i*128 + j, 0..16383
    const unsigned i   = row >> 7;
    const unsigned j   = row & 127u;
    const unsigned k   = i + j + 1u;          // 1..255

    // Wave-uniform branchless base select in integer domain (s_cselect_b64):
    // bias the wise base by -128 rows so addr = base + k*ROW_BYTES either way.
    const uintptr_t pbase = (uintptr_t)prev_;
    const uintptr_t wbase = (uintptr_t)wise_ - (uintptr_t)CRTN_NEI * ROW_BYTES;
    const uintptr_t sbase = ((k < CRTN_NEI) ? pbase : wbase) + (uintptr_t)k * ROW_BYTES;

    const unsigned t = threadIdx.x;

    // Cast final addresses into explicit global address space -> global_* ops.
    const gv4f* __restrict__ src = (const gv4f*)sbase + t;
    gv4f* __restrict__       dst = (gv4f*)((uintptr_t)out_ + (uintptr_t)row * ROW_BYTES) + t;

    // Stage the whole 64 KB row slice in registers: 16 global_load_b128 in
    // flight per wave (single load clause, descending s_wait_loadcnt).
    v4f buf[VEC_PER_THREAD];
    #pragma unroll
    for (int it = 0; it < VEC_PER_THREAD; ++it)
        buf[it] = src[it * BLOCK_THREADS];          // offset = it*4096 bytes (imm)

    // Stream out with non-temporal hint: coalesced 4 KB bursts per iteration.
    #pragma unroll
    for (int it = 0; it < VEC_PER_THREAD; ++it)
        __builtin_nontemporal_store(buf[it], &dst[it * BLOCK_THREADS]);
}

extern "C" void kernel_launch(void* const* d_in, const int* in_sizes, int n_in,
                              void* d_out, int out_size, void* d_ws, size_t ws_size,
                              hipStream_t stream) {
    (void)in_sizes; (void)n_in; (void)out_size; (void)d_ws; (void)ws_size;

    const float* neighbor_mem = (const float*)d_in[0];  // (13, 128, 16, 1024)
    const float* wise_inputs  = (const float*)d_in[1];  // (128, 16, 1024)
    float*       out          = (float*)d_out;          // (128, 128, 16, 1024)

    // Last memory layer: neighbor_mem[12]
    const float* prev = neighbor_mem + (size_t)(CRTN_LAYERS - 1) * CRTN_NEI * ROW_FLOATS;

    dim3 grid(CRTN_S * CRTN_S);               // 16384 blocks, one output row each
    dim3 block(BLOCK_THREADS);
    crtn_query_gather<<<grid, block, 0, stream>>>(
        (const v4f*)prev, (const v4f*)wise_inputs, (v4f*)out);
}